// GCNEncoder_49237505081833
// MI455X (gfx1250) — compile-verified
//
#include <hip/hip_runtime.h>
#include <cstdint>

typedef __attribute__((ext_vector_type(2))) float v2f;
typedef __attribute__((ext_vector_type(8))) float v8f;

static constexpr int IN_DIM = 256;
static constexpr int HID    = 64;
static constexpr int EMB    = 10;

// ---------------------------------------------------------------------------
// Utility kernels
// ---------------------------------------------------------------------------
__global__ void fill_kernel(float* __restrict__ p, float v, long long n) {
  long long i = (long long)blockIdx.x * blockDim.x + threadIdx.x;
  if (i < n) p[i] = v;
}

__global__ void deg_kernel(const int* __restrict__ dst, float* __restrict__ deg, int e) {
  int i = blockIdx.x * blockDim.x + threadIdx.x;
  if (i < e) atomicAdd(&deg[dst[i]], 1.0f);
}

__global__ void rsqrt_kernel(float* __restrict__ p, int n) {
  int i = blockIdx.x * blockDim.x + threadIdx.x;
  if (i < n) p[i] = rsqrtf(p[i]);
}

// ---------------------------------------------------------------------------
// WMMA GEMM: H[nrows, kout] = X[nrows, KIN] @ W[KIN, kout]
// 256 threads = 8 wave32; each wave owns a 16-row x (NTILES*16)-col tile.
//  * X tile staged per-wave into LDS via gfx1250 async-to-LDS b128 loads
//    (ASYNCcnt), rows padded to 68 floats -> conflict-free ds_load_b64
//    A-fragments (banks (4m+2hi) mod 64 disjoint).
//  * W staged in 64-row chunks, pre-swizzled into WMMA B-fragment order so
//    each lane reads one aligned ds_load_b64 straight into the wmma operand.
//  * V_WMMA_F32_16X16X4_F32, exact fp32 accumulate.
// ---------------------------------------------------------------------------
template <int KIN, int NTILES>
__global__ __launch_bounds__(256) void gemm_wmma_kernel(
    const float* __restrict__ X, const float* __restrict__ W,
    float* __restrict__ H, int nrows, int kout) {
  constexpr int XPITCH = 68;                     // padded row (floats)
  __shared__ float sX[8][16 * XPITCH];           // per-wave 16x64 X tile
  __shared__ float sW[16 * NTILES * 64];         // [kg][t][lane][2] fragments

  const int wave = threadIdx.x >> 5;
  const int lane = threadIdx.x & 31;
  const int m    = lane & 15;   // A row / B col within tile
  const int hi   = lane >> 4;   // half-wave K-pair select
  const int row0 = (blockIdx.x * 8 + wave) * 16;

  v8f acc[NTILES] = {};

  for (int kc = 0; kc < KIN; kc += 64) {
    // ---- stage this wave's 16x64 X chunk via async-to-LDS (ASYNCcnt) ----
#pragma unroll
    for (int j = 0; j < 8; ++j) {
      int idx = j * 32 + lane;                   // 0..255 float4 slots
      int r   = idx >> 4;                        // tile row 0..15
      int c4  = idx & 15;                        // float4 col 0..15
      int gr  = row0 + r;
      if (gr >= nrows) gr = nrows - 1;           // clamp: garbage rows never stored
      const float* gp = X + (size_t)gr * KIN + kc + c4 * 4;
      unsigned ldsb = (unsigned)(uintptr_t)&sX[wave][r * XPITCH + c4 * 4];
      asm volatile("global_load_async_to_lds_b128 %0, %1, off"
                   :: "v"(ldsb), "v"(gp) : "memory");
    }

    // ---- stage W chunk, swizzled into B-fragment order ----
    __syncthreads();                             // protect sW reuse
    for (int i = threadIdx.x; i < 16 * NTILES * 32; i += 256) {
      int kg = i / (NTILES * 32);
      int rm = i % (NTILES * 32);
      int t  = rm >> 5;
      int l  = rm & 31;
      int lm = l & 15, lh = l >> 4;
      int grow = kc + kg * 4 + 2 * lh;
      int col  = t * 16 + lm;
      float w0 = 0.0f, w1 = 0.0f;
      if (col < kout) {
        w0 = W[(size_t)grow * kout + col];
        w1 = W[(size_t)(grow + 1) * kout + col];
      }
      sW[i * 2 + 0] = w0;
      sW[i * 2 + 1] = w1;
    }
    __syncthreads();

    // ---- drain async X tile for this wave ----
    asm volatile("s_wait_asynccnt 0x0" ::: "memory");

    // ---- 16 WMMA k-steps over the 64-wide chunk ----
#pragma unroll 4
    for (int kg = 0; kg < 16; ++kg) {
      v2f a = *(const v2f*)&sX[wave][m * XPITCH + kg * 4 + 2 * hi];
#pragma unroll
      for (int t = 0; t < NTILES; ++t) {
        v2f b = *(const v2f*)&sW[((kg * NTILES + t) * 32 + lane) * 2];
        acc[t] = __builtin_amdgcn_wmma_f32_16x16x4_f32(
            false, a, false, b, (short)0, acc[t], false, false);
      }
    }
  }

  // C/D layout: VGPR r -> (M = r + 8*hi, N = m)
#pragma unroll
  for (int t = 0; t < NTILES; ++t) {
#pragma unroll
    for (int r = 0; r < 8; ++r) {
      const int orow = row0 + r + 8 * hi;
      const int ocol = t * 16 + m;
      if (orow < nrows && ocol < kout)
        H[(size_t)orow * kout + ocol] = acc[t][r];
    }
  }
}

// ---------------------------------------------------------------------------
// Edge scatter, F=64: 16 threads per edge, float4 per thread.
// agg[dst] += h[src] * dinv[src]*dinv[dst]
// ---------------------------------------------------------------------------
__global__ void scatter64_kernel(const int* __restrict__ src, const int* __restrict__ dst,
                                 const float* __restrict__ dinv, const float* __restrict__ h,
                                 float* __restrict__ agg, int e) {
  long long tid = (long long)blockIdx.x * blockDim.x + threadIdx.x;
  long long ei  = tid >> 4;
  int c         = (int)(tid & 15);
  if (ei >= e) return;
  int s = src[ei], d = dst[ei];
  float w = dinv[s] * dinv[d];
  float4 v = *(const float4*)(h + (size_t)s * 64 + c * 4);
  float* ap = agg + (size_t)d * 64 + c * 4;
  atomicAdd(ap + 0, v.x * w);
  atomicAdd(ap + 1, v.y * w);
  atomicAdd(ap + 2, v.z * w);
  atomicAdd(ap + 3, v.w * w);
}

// Edge scatter, F=10: 16 threads per edge (6 idle), scalar atomics.
__global__ void scatter10_kernel(const int* __restrict__ src, const int* __restrict__ dst,
                                 const float* __restrict__ dinv, const float* __restrict__ h,
                                 float* __restrict__ agg, int e) {
  long long tid = (long long)blockIdx.x * blockDim.x + threadIdx.x;
  long long ei  = tid >> 4;
  int c         = (int)(tid & 15);
  if (ei >= e || c >= EMB) return;
  int s = src[ei], d = dst[ei];
  float w = dinv[s] * dinv[d];
  atomicAdd(&agg[(size_t)d * EMB + c], h[(size_t)s * EMB + c] * w);
}

// ---------------------------------------------------------------------------
// out = [relu](agg + h * dinv^2 + b)
// ---------------------------------------------------------------------------
template <int F, bool RELU>
__global__ void combine_kernel(const float* __restrict__ agg, const float* __restrict__ h,
                               const float* __restrict__ dinv, const float* __restrict__ b,
                               float* __restrict__ out, int n) {
  long long tid = (long long)blockIdx.x * blockDim.x + threadIdx.x;
  if (tid >= (long long)n * F) return;
  int i = (int)(tid / F);
  int f = (int)(tid % F);
  float di = dinv[i];
  float v = agg[tid] + h[tid] * di * di + b[f];
  out[tid] = RELU ? fmaxf(v, 0.0f) : v;
}

// ---------------------------------------------------------------------------
// Launcher
// ---------------------------------------------------------------------------
static inline unsigned nblk(long long work, int bs) {
  return (unsigned)((work + bs - 1) / bs);
}

extern "C" void kernel_launch(void* const* d_in, const int* in_sizes, int n_in,
                              void* d_out, int out_size, void* d_ws, size_t ws_size,
                              hipStream_t stream) {
  (void)n_in; (void)out_size; (void)ws_size;

  const float* x  = (const float*)d_in[0];
  const int*   ei = (const int*)d_in[1];
  const float* W1 = (const float*)d_in[2];
  const float* b1 = (const float*)d_in[3];
  const float* W2 = (const float*)d_in[4];
  const float* b2 = (const float*)d_in[5];
  const float* W3 = (const float*)d_in[6];
  const float* b3 = (const float*)d_in[7];
  float* out = (float*)d_out;

  const int n = in_sizes[0] / IN_DIM;   // 100000 nodes
  const int e = in_sizes[1] / 2;        // 3.2M edges
  const int* src = ei;
  const int* dst = ei + e;

  // Workspace layout: dinv[n] | bufA[n*64] | bufB[n*64]  (~51.6 MB)
  char* ws = (char*)d_ws;
  float* dinv = (float*)ws;
  size_t ofs  = (((size_t)n * 4) + 255) & ~(size_t)255;
  float* bufA = (float*)(ws + ofs);
  float* bufB = (float*)(ws + ofs + (size_t)n * HID * 4);

  const int BS = 256;
  const long long nh = (long long)n * HID;
  const long long ne = (long long)n * EMB;

  // --- degree / symmetric norm (shared by all layers) ---
  fill_kernel<<<nblk(n, BS), BS, 0, stream>>>(dinv, 1.0f, n);           // self-loop
  deg_kernel<<<nblk(e, BS), BS, 0, stream>>>(dst, dinv, e);
  rsqrt_kernel<<<nblk(n, BS), BS, 0, stream>>>(dinv, n);

  // --- layer 1: 256 -> 64, ReLU ---
  gemm_wmma_kernel<IN_DIM, 4><<<nblk(n, 128), 256, 0, stream>>>(x, W1, bufA, n, HID);
  fill_kernel<<<nblk(nh, BS), BS, 0, stream>>>(bufB, 0.0f, nh);
  scatter64_kernel<<<nblk((long long)e * 16, BS), BS, 0, stream>>>(src, dst, dinv, bufA, bufB, e);
  combine_kernel<HID, true><<<nblk(nh, BS), BS, 0, stream>>>(bufB, bufA, dinv, b1, bufA, n);

  // --- layer 2: 64 -> 64, ReLU ---
  gemm_wmma_kernel<HID, 4><<<nblk(n, 128), 256, 0, stream>>>(bufA, W2, bufB, n, HID);
  fill_kernel<<<nblk(nh, BS), BS, 0, stream>>>(bufA, 0.0f, nh);
  scatter64_kernel<<<nblk((long long)e * 16, BS), BS, 0, stream>>>(src, dst, dinv, bufB, bufA, e);
  combine_kernel<HID, true><<<nblk(nh, BS), BS, 0, stream>>>(bufA, bufB, dinv, b2, bufB, n);

  // --- layer 3: 64 -> 10, linear ---
  gemm_wmma_kernel<HID, 1><<<nblk(n, 128), 256, 0, stream>>>(bufB, W3, bufA, n, EMB);
  fill_kernel<<<nblk(ne, BS), BS, 0, stream>>>(out, 0.0f, ne);
  scatter10_kernel<<<nblk((long long)e * 16, BS), BS, 0, stream>>>(src, dst, dinv, bufA, out, e);
  combine_kernel<EMB, false><<<nblk(ne, BS), BS, 0, stream>>>(out, bufA, dinv, b3, out, n);
}